// VariationalMonteCarloRBM_79714593013768
// MI455X (gfx1250) — compile-verified
//
#include <hip/hip_runtime.h>
#include <hip/hip_bf16.h>

typedef __attribute__((ext_vector_type(2))) float v2f;
typedef __attribute__((ext_vector_type(8))) float v8f;

#define B_TOT 4096
#define N_SPIN 64
#define H_HID 128
#define WPAD 129          // padded LDS row stride: (n*129+h)%64 distinct per n -> no bank conflicts
#define ROWS_PER_BLK 16

// log(cosh(x)) = |x| + log(1 + exp(-2|x|)) - log(2)
__device__ __forceinline__ float logcoshf_fast(float x) {
    float ax = fabsf(x);
    float e  = __expf(-2.0f * ax);          // v_exp_f32
    return ax + __logf(1.0f + e) - 0.69314718055994530942f;  // v_log_f32
}

__global__ __launch_bounds__(256) void rbm_eloc_kernel(
    const float* __restrict__ samples,  // (4096, 64), values +-1
    const float* __restrict__ W,        // (64, 128)
    const float* __restrict__ bv,       // (64)
    const float* __restrict__ bh,       // (128)
    float* __restrict__ out)            // (2, 4096) flat: [real(4096), imag(4096)]
{
    __shared__ float sW[N_SPIN * WPAD];          // 64 x 129
    __shared__ float sTheta[ROWS_PER_BLK * WPAD];// 16 x 129
    __shared__ float sS[ROWS_PER_BLK * N_SPIN];  // 16 x 64
    __shared__ float sFlip[ROWS_PER_BLK * N_SPIN];
    __shared__ float sLogPartial[256];
    __shared__ float sBh[H_HID];
    __shared__ float sBv[N_SPIN];
    __shared__ float sLogSum[ROWS_PER_BLK];
    __shared__ float sZZ[ROWS_PER_BLK];

    const int tid  = threadIdx.x;
    const int lane = tid & 31;
    const int wave = tid >> 5;
    const int row0 = blockIdx.x * ROWS_PER_BLK;

    // ---- Stage W (padded), bh, bv, and the 16x64 sample tile into LDS ----
    for (int i = tid; i < N_SPIN * H_HID; i += 256) {
        int n = i >> 7, h = i & 127;
        sW[n * WPAD + h] = W[i];
    }
    if (tid < H_HID) sBh[tid] = bh[tid];
    if (tid < N_SPIN) sBv[tid] = bv[tid];
    for (int i = tid; i < ROWS_PER_BLK * N_SPIN; i += 256) {
        sS[i] = samples[row0 * N_SPIN + i];
    }
    __syncthreads();

    // ---- theta = S(16x64) @ W(64x128) + bh via V_WMMA_F32_16X16X4_F32 ----
    // Wave w computes output columns [16w, 16w+16). 16 K-steps of K=4.
    // 32-bit A 16x4 layout: lanes 0-15 -> K={0,1} in v[0],v[1]; lanes 16-31 -> K={2,3}.
    // 32-bit C/D layout: lane in [0,16): N=lane, v[j] = M=j; lane in [16,32): M=8+j.
    {
        const int m     = lane & 15;
        const int khalf = (lane >> 4) << 1;   // 0 or 2
        const int c0    = wave << 4;          // column base for this wave
        v8f c = {};
        #pragma unroll
        for (int kt = 0; kt < 16; ++kt) {
            const int k0 = (kt << 2) + khalf;
            v2f a, b;
            a.x = sS[m * N_SPIN + k0];
            a.y = sS[m * N_SPIN + k0 + 1];
            b.x = sW[k0 * WPAD + c0 + m];
            b.y = sW[(k0 + 1) * WPAD + c0 + m];
            c = __builtin_amdgcn_wmma_f32_16x16x4_f32(
                    /*neg_a=*/false, a, /*neg_b=*/false, b,
                    /*c_mod=*/(short)0, c, /*reuse_a=*/false, /*reuse_b=*/false);
        }
        const int rbase = (lane >> 4) << 3;   // 0 or 8
        #pragma unroll
        for (int j = 0; j < 8; ++j)
            sTheta[(rbase + j) * WPAD + c0 + m] = c[j] + sBh[c0 + m];
    }
    __syncthreads();

    // ---- zz per row (nearest-neighbor Ising term with periodic roll) ----
    if (tid < ROWS_PER_BLK) {
        float zz = 0.0f;
        for (int n = 0; n < N_SPIN; ++n)
            zz += sS[tid * N_SPIN + n] * sS[tid * N_SPIN + ((n + 1) & 63)];
        sZZ[tid] = zz;
    }

    // ---- logsum[r] = sum_h logcosh(theta[r][h]) : 16 partial lanes per row ----
    {
        const int r = tid >> 4, hc = tid & 15;
        float acc = 0.0f;
        #pragma unroll
        for (int h = hc; h < H_HID; h += 16)
            acc += logcoshf_fast(sTheta[r * WPAD + h]);
        sLogPartial[tid] = acc;
    }
    __syncthreads();
    if (tid < ROWS_PER_BLK) {
        float s = 0.0f;
        #pragma unroll
        for (int j = 0; j < 16; ++j) s += sLogPartial[tid * 16 + j];
        sLogSum[tid] = s;
    }
    __syncthreads();

    // ---- Main flip loop: 16*64 tasks; task = r*64 + n ----
    // Within a wave all lanes share r (broadcast theta reads); n distinct
    // (conflict-free W reads thanks to WPAD=129).
    for (int t = tid; t < ROWS_PER_BLK * N_SPIN; t += 256) {
        const int r = t >> 6, n = t & 63;
        const float coef = -2.0f * sS[r * N_SPIN + n];
        const float* __restrict__ wrow = &sW[n * WPAD];
        const float* __restrict__ trow = &sTheta[r * WPAD];
        float acc = 0.0f;
        #pragma unroll 4
        for (int h = 0; h < H_HID; ++h)
            acc += logcoshf_fast(fmaf(coef, wrow[h], trow[h]));
        sFlip[t] = acc;
    }
    __syncthreads();

    // ---- exp_term and energy accumulation; ln_bias cancels except -s*bv ----
    if (tid < ROWS_PER_BLK) {
        const int r = tid;
        const float logsum = sLogSum[r];
        float sumR = 0.0f, sumI = 0.0f;
        for (int n = 0; n < N_SPIN; ++n) {
            const float s    = sS[r * N_SPIN + n];
            const float expo = 0.5f * (sFlip[r * N_SPIN + n] - logsum) - s * sBv[n];
            const float e    = __expf(expo);
            sumR += e;
            sumI += (n & 1) ? e : -e;     // sign[n] = -1 for even n
        }
        const int b = row0 + r;
        out[b]         = -sZZ[r] - sumR;  // -JZ*zz - G_REAL*sum   (JZ=G_REAL=1)
        out[B_TOT + b] = 0.5f * sumI;     // G_IMAG * sum          (G_IMAG=0.5)
    }
}

extern "C" void kernel_launch(void* const* d_in, const int* in_sizes, int n_in,
                              void* d_out, int out_size, void* d_ws, size_t ws_size,
                              hipStream_t stream) {
    (void)in_sizes; (void)n_in; (void)d_ws; (void)ws_size; (void)out_size;
    const float* samples = (const float*)d_in[0];  // (4096, 64)
    const float* W       = (const float*)d_in[1];  // (64, 128)
    const float* bv      = (const float*)d_in[2];  // (1, 64)
    const float* bh      = (const float*)d_in[3];  // (1, 128)
    float* out           = (float*)d_out;          // (2, 4096)

    dim3 grid(B_TOT / ROWS_PER_BLK);  // 256 workgroups
    dim3 block(256);                  // 8 wave32 per workgroup
    rbm_eloc_kernel<<<grid, block, 0, stream>>>(samples, W, bv, bh, out);
}